// Attention_7911329759249
// MI455X (gfx1250) — compile-verified
//
#include <hip/hip_runtime.h>
#include <hip/hip_bf16.h>
#include <stdint.h>

typedef float v2f __attribute__((ext_vector_type(2)));
typedef float v8f __attribute__((ext_vector_type(8)));
typedef unsigned int v4u __attribute__((ext_vector_type(4)));
typedef int v4i __attribute__((ext_vector_type(4)));
typedef int v8i __attribute__((ext_vector_type(8)));

namespace {
constexpr int kB = 32, kS = 4096, kH = 1024;
constexpr int kChunks = 16;             // s-chunks per batch
constexpr int kSC = kS / kChunks;       // 256 rows per block
constexpr int kTS = 16;                 // rows per LDS tile (one WMMA M-tile)
constexpr int kNT = kSC / kTS;          // 16 tiles per block
// TDM pads 2 dwords every 256 data dwords -> padded row stride in dwords:
constexpr int kRowQ = kH + 2 * (kH / 256);   // 1032
constexpr int kBufQ = kTS * kRowQ;           // 16512 dwords = 64.5 KB

// dynamic-LDS layout (float offsets)
constexpr int SM_BUF0  = 0;
constexpr int SM_BUF1  = kBufQ;
constexpr int SM_DEC   = 2 * kBufQ;          // decoder state, 1024 f32
constexpr int SM_SPART = SM_DEC + kH;        // 8 waves * 16 partial scores
constexpr int SM_SCR   = SM_SPART + 128;     // 16 reduced scores
constexpr int SM_WEXP  = SM_SCR + 16;        // 16 exp-weights
constexpr int SM_STATE = SM_WEXP + 16;       // [0]=m [1]=l [2]=corr
constexpr int SM_FLOATS = SM_STATE + 4;      // 34212 floats ~= 137 KB

// workspace layout (float offsets)
constexpr size_t WS_SCORES = 0;                                       // B*S raw scores
constexpr size_t WS_CPART  = (size_t)kB * kS;                         // B*Chunks*H ctx partials
constexpr size_t WS_M      = WS_CPART + (size_t)kB * kChunks * kH;
constexpr size_t WS_L      = WS_M + (size_t)kB * kChunks;
}

// One TDM descriptor moves a full 16 x 1024 fp32 tile Global->LDS with padding.
__device__ __forceinline__ void tdm_load_tile(const float* gsrc, unsigned lds_byte) {
  unsigned long long ga = (unsigned long long)(uintptr_t)gsrc;
  // D# group0: count=1, lds_addr, 57-bit global addr, type=2
  v4u g0 = { 1u,
             lds_byte,
             (unsigned)(ga & 0xFFFFFFFFull),
             (unsigned)((ga >> 32) & 0x01FFFFFFull) | (2u << 30) };
  // D# group1: data_size=4B, pad_enable, pad_interval=256dw(code 7), pad_amount=2dw(code 1)
  int w0 = (2 << 16) | (1 << 20) | (7 << 22) | (1 << 25);
  v8i g1 = { w0,
             (int)((unsigned)kH << 16),   // tensor_dim0 = 1024 (low16 of bits 79:48)
             (int)((unsigned)kS << 16),   // tensor_dim1 = 4096 (low16), dim0 hi16 = 0
             (int)((unsigned)kH << 16),   // tile_dim0 = 1024, dim1 hi16 = 0
             kTS,                         // tile_dim1 = 16, tile_dim2 = 0
             kH,                          // tensor_dim0_stride = 1024 (low32)
             0, 0 };
  v4i z4 = { 0, 0, 0, 0 };
  v8i z8 = { 0, 0, 0, 0, 0, 0, 0, 0 };
  __builtin_amdgcn_tensor_load_to_lds(g0, g1, z4, z4, z8, 0);
}

__global__ void attn_fused_pass(const float* __restrict__ dec,
                                const float* __restrict__ enc,
                                float* __restrict__ ws) {
  extern __shared__ float smem[];
  const int chunk = blockIdx.x;
  const int b = blockIdx.y;
  const int tid = threadIdx.x;
  const int lane = tid & 31;
  const int wv = tid >> 5;

  // stage decoder state b into LDS (coalesced float4)
  {
    const float4 v = *(const float4*)(dec + (size_t)b * kH + tid * 4);
    *(float4*)&smem[SM_DEC + tid * 4] = v;
  }
  if (tid == 0) { smem[SM_STATE + 0] = -__builtin_inff(); smem[SM_STATE + 1] = 0.f; }

  const float* gbase = enc + ((size_t)b * kS + (size_t)chunk * kSC) * kH;
  const unsigned lds_base = (unsigned)(uintptr_t)(&smem[0]); // low 32 bits = LDS byte addr

  if (wv == 0) {
    tdm_load_tile(gbase, lds_base + SM_BUF0 * 4u);
    __builtin_amdgcn_s_wait_tensorcnt(0);
  }
  __syncthreads();

  float cacc[4] = {0.f, 0.f, 0.f, 0.f};   // this thread's 4 h-columns of running context
  const int ct = tid * 4;
  const int r = lane & 15;
  const int koff = (lane >> 4) * 2;

  // Per-thread constant LDS sub-offsets. Within a wave's 128-wide k-range the
  // TDM pad term 2*(c>>8) is the constant 2*(wv>>1), so all addresses are
  // linear in kk / j -> single base VGPR + immediate DS offsets, no spills.
  const int aOff = r * kRowQ + wv * 128 + koff + 2 * (wv >> 1); // A-fragment base (per buffer)
  const int bOff = SM_DEC + wv * 128 + koff;                    // B-fragment base
  const int cOff = ct + 2 * (ct >> 8);                          // context-column base (per buffer)
  const float* const bp = &smem[bOff];

  float* wsScore = ws + WS_SCORES + (size_t)b * kS + (size_t)chunk * kSC;

  for (int t = 0; t < kNT; ++t) {
    const int cur = (t & 1) ? SM_BUF1 : SM_BUF0;
    if ((t + 1) < kNT && wv == 0) {
      const int nxt = ((t + 1) & 1) ? SM_BUF1 : SM_BUF0;
      tdm_load_tile(gbase + (size_t)(t + 1) * kTS * kH, lds_base + (unsigned)nxt * 4u);
    }

    // ---- scores for 16 rows via V_WMMA_F32_16X16X4_F32 (wave w owns k in [w*128, w*128+128)) ----
    const float* const ap = &smem[cur + aOff];
    v8f acc = {0.f, 0.f, 0.f, 0.f, 0.f, 0.f, 0.f, 0.f};
#pragma unroll 8
    for (int kk = 0; kk < 32; ++kk) {
      v2f a  = *(const v2f*)(ap + kk * 4);   // A[r][k0+koff .. +1]
      v2f bf = *(const v2f*)(bp + kk * 4);   // B[k][n] = dec[k0+k] (all columns equal)
      acc = __builtin_amdgcn_wmma_f32_16x16x4_f32(false, a, false, bf, (short)0, acc,
                                                  false, false);
    }
    // column 0 of D: lane 0 holds M=0..7 in acc[0..7], lane 16 holds M=8..15
    if (lane == 0 || lane == 16) {
      const int base = SM_SPART + wv * 16 + (lane >> 4) * 8;
#pragma unroll
      for (int j = 0; j < 8; ++j) smem[base + j] = acc[j];
    }
    __syncthreads();
    if (tid < 16) {
      float s = 0.f;
#pragma unroll
      for (int w = 0; w < 8; ++w) s += smem[SM_SPART + w * 16 + tid];
      smem[SM_SCR + tid] = s;
      wsScore[t * kTS + tid] = s;     // raw score for final attn computation
    }
    __syncthreads();

    // ---- online softmax state update (block-wide scalars) ----
    if (tid == 0) {
      const float mold = smem[SM_STATE + 0];
      float mnew = mold;
#pragma unroll
      for (int j = 0; j < 16; ++j) mnew = fmaxf(mnew, smem[SM_SCR + j]);
      const float corr = __expf(mold - mnew);
      float lsum = 0.f;
#pragma unroll
      for (int j = 0; j < 16; ++j) {
        const float wj = __expf(smem[SM_SCR + j] - mnew);
        smem[SM_WEXP + j] = wj;
        lsum += wj;
      }
      smem[SM_STATE + 1] = smem[SM_STATE + 1] * corr + lsum;
      smem[SM_STATE + 0] = mnew;
      smem[SM_STATE + 2] = corr;
    }
    __syncthreads();

    // ---- context accumulation: thread owns 4 h-columns ----
    const float corr = smem[SM_STATE + 2];
#pragma unroll
    for (int e = 0; e < 4; ++e) cacc[e] *= corr;
    const float* const tp = &smem[cur + cOff];
#pragma unroll
    for (int j = 0; j < kTS; ++j) {
      const float wj = smem[SM_WEXP + j];
      v2f x0 = *(const v2f*)(tp + j * kRowQ);
      v2f x1 = *(const v2f*)(tp + j * kRowQ + 2);
      cacc[0] += wj * x0.x; cacc[1] += wj * x0.y;
      cacc[2] += wj * x1.x; cacc[3] += wj * x1.y;
    }
    __syncthreads();                                    // everyone done with buffers
    if (wv == 0) __builtin_amdgcn_s_wait_tensorcnt(0);  // next tile's DMA complete
    __syncthreads();                                    // release
  }

  float* cp = ws + WS_CPART + ((size_t)b * kChunks + chunk) * kH + ct;
#pragma unroll
  for (int e = 0; e < 4; ++e) cp[e] = cacc[e];
  if (tid == 0) {
    ws[WS_M + (size_t)b * kChunks + chunk] = smem[SM_STATE + 0];
    ws[WS_L + (size_t)b * kChunks + chunk] = smem[SM_STATE + 1];
  }
}

__global__ void attn_combine(const float* __restrict__ ws, float* __restrict__ out) {
  const int b = blockIdx.x;
  const int tid = threadIdx.x;
  const float* mP = ws + WS_M + (size_t)b * kChunks;
  const float* lP = ws + WS_L + (size_t)b * kChunks;

  float M = -__builtin_inff();
#pragma unroll
  for (int i = 0; i < kChunks; ++i) M = fmaxf(M, mP[i]);
  float L = 0.f;
#pragma unroll
  for (int i = 0; i < kChunks; ++i) L += lP[i] * __expf(mP[i] - M);
  const float invL = 1.f / L;

  // context: merge chunk partials (thread owns 4 h-columns)
  const int ct = tid * 4;
  float acc[4] = {0.f, 0.f, 0.f, 0.f};
  for (int i = 0; i < kChunks; ++i) {
    const float sc = __expf(mP[i] - M);
    const float* cp = ws + WS_CPART + ((size_t)b * kChunks + i) * kH + ct;
#pragma unroll
    for (int e = 0; e < 4; ++e) acc[e] += sc * cp[e];
  }
  float* ctx = out + (size_t)b * kH + ct;
#pragma unroll
  for (int e = 0; e < 4; ++e) ctx[e] = acc[e] * invL;

  // attn = exp(score - M) / L
  const float* sc = ws + WS_SCORES + (size_t)b * kS;
  float* attn = out + (size_t)kB * kH + (size_t)b * kS;
  for (int s = tid; s < kS; s += 256) attn[s] = __expf(sc[s] - M) * invL;
}

extern "C" void kernel_launch(void* const* d_in, const int* in_sizes, int n_in,
                              void* d_out, int out_size, void* d_ws, size_t ws_size,
                              hipStream_t stream) {
  (void)in_sizes; (void)n_in; (void)out_size; (void)ws_size;
  const float* dec = (const float*)d_in[0];
  const float* enc = (const float*)d_in[1];
  float* ws = (float*)d_ws;
  float* out = (float*)d_out;

  dim3 gridA(kChunks, kB);
  const size_t smemBytes = (size_t)SM_FLOATS * sizeof(float);
  attn_fused_pass<<<gridA, 256, smemBytes, stream>>>(dec, enc, ws);
  attn_combine<<<dim3(kB), 256, 0, stream>>>(ws, out);
}